// CARAFE_72275709657323
// MI455X (gfx1250) — compile-verified
//
#include <hip/hip_runtime.h>
#include <math.h>

typedef float v2f __attribute__((ext_vector_type(2)));
typedef float v8f __attribute__((ext_vector_type(8)));

#define BN_EPS 1e-5f

// ------------------------------------------------------------------
// Kernel 1: 1x1 conv (256 -> 64) + BN + ReLU as f32 WMMA GEMM
//   M = B*H*W = 8192, N = 64, K = 256, tile 16x16 per wave
// ------------------------------------------------------------------
__global__ __launch_bounds__(256) void k1_comp(
    const float* __restrict__ X,      // (2,256,64,64)
    const float* __restrict__ Wc,     // (64,256)
    const float* __restrict__ bc,
    const float* __restrict__ g1, const float* __restrict__ be1,
    const float* __restrict__ mu1, const float* __restrict__ va1,
    float* __restrict__ Y1)           // (2,64,64,64)
{
    const int lane = threadIdx.x & 31;
    const int wave = threadIdx.x >> 5;
    const int gw = blockIdx.x * 8 + wave;          // 0..2047
    const int mt = gw >> 2;                        // 0..511
    const int nt = gw & 3;                         // 0..3
    const int m0 = mt * 16;
    const int n0 = nt * 16;

    const int mrow  = m0 + (lane & 15);
    const int bimg  = mrow >> 12;                  // / 4096
    const int hw    = mrow & 4095;
    const int khalf = (lane >> 4) * 2;             // 0 or 2
    const int ncol  = n0 + (lane & 15);

    const float* Xb = X + bimg * (256 * 4096) + hw;
    const float* Wn = Wc + ncol * 256;

    v8f acc = {};
    for (int kk = 0; kk < 256; kk += 4) {
        const int ka = kk + khalf;
        v2f a, b;
        a.x = Xb[(ka    ) * 4096];
        a.y = Xb[(ka + 1) * 4096];
        b.x = Wn[ka];
        b.y = Wn[ka + 1];
        acc = __builtin_amdgcn_wmma_f32_16x16x4_f32(false, a, false, b,
                                                    (short)0, acc, false, false);
    }

    // folded BN1 + bias + ReLU epilogue
    const float s1 = g1[ncol] * rsqrtf(va1[ncol] + BN_EPS);
    const float t1 = s1 * (bc[ncol] - mu1[ncol]) + be1[ncol];
    const int rofs = (lane >> 4) * 8;
    #pragma unroll
    for (int r = 0; r < 8; ++r) {
        const int m = m0 + r + rofs;
        float v = s1 * acc[r] + t1;
        v = v > 0.f ? v : 0.f;
        const int bi = m >> 12, phw = m & 4095;
        Y1[bi * (64 * 4096) + ncol * 4096 + phw] = v;
    }
}

// ------------------------------------------------------------------
// Kernel 2: 3x3 conv (64 -> 100, pad 1) + BN as f32 WMMA GEMM
//   M = 8192, N = 112 (100 padded to 7 tiles), K = 576, im2col on the fly
// ------------------------------------------------------------------
__global__ __launch_bounds__(256) void k2_enc(
    const float* __restrict__ Y1,     // (2,64,64,64)
    const float* __restrict__ We,     // (100,576)
    const float* __restrict__ bev,
    const float* __restrict__ g2, const float* __restrict__ be2,
    const float* __restrict__ mu2, const float* __restrict__ va2,
    float* __restrict__ Z)            // (2,100,64,64)
{
    const int lane = threadIdx.x & 31;
    const int wave = threadIdx.x >> 5;
    const int gw = blockIdx.x * 8 + wave;          // 0..3583
    const int mt = gw / 7;
    const int nt = gw % 7;
    const int m0 = mt * 16;
    const int n0 = nt * 16;

    const int mrow  = m0 + (lane & 15);
    const int bimg  = mrow >> 12;
    const int hw    = mrow & 4095;
    const int h     = hw >> 6, w = hw & 63;
    const int khalf = (lane >> 4) * 2;
    const int ncol  = n0 + (lane & 15);
    const bool nok  = ncol < 100;

    const float* Yb = Y1 + bimg * (64 * 4096);

    v8f acc = {};
    for (int kk = 0; kk < 576; kk += 4) {
        v2f a, b;
        #pragma unroll
        for (int u = 0; u < 2; ++u) {
            const int k  = kk + khalf + u;
            const int c  = k / 9;
            const int off = k - c * 9;
            const int hh = h + off / 3 - 1;
            const int ww = w + off % 3 - 1;
            float av = 0.f;
            if ((unsigned)hh < 64u && (unsigned)ww < 64u)
                av = Yb[c * 4096 + hh * 64 + ww];
            if (u == 0) a.x = av; else a.y = av;
        }
        const int kb = kk + khalf;
        b.x = nok ? We[ncol * 576 + kb]     : 0.f;
        b.y = nok ? We[ncol * 576 + kb + 1] : 0.f;
        acc = __builtin_amdgcn_wmma_f32_16x16x4_f32(false, a, false, b,
                                                    (short)0, acc, false, false);
    }

    float s2 = 0.f, t2 = 0.f;
    if (nok) {
        s2 = g2[ncol] * rsqrtf(va2[ncol] + BN_EPS);
        t2 = s2 * (bev[ncol] - mu2[ncol]) + be2[ncol];
    }
    const int rofs = (lane >> 4) * 8;
    #pragma unroll
    for (int r = 0; r < 8; ++r) {
        if (nok) {
            const int m = m0 + r + rofs;
            const int bi = m >> 12, phw = m & 4095;
            Z[bi * (100 * 4096) + ncol * 4096 + phw] = s2 * acc[r] + t2;
        }
    }
}

// ------------------------------------------------------------------
// Kernel 3: PixelShuffle(2) + softmax(25) + 5x5 dilated reassembly.
//   Block = 8x8 low-res tile; thread = (lowres pixel, subpixel).
//   X read through a 12x12 zero-padded LDS halo tile, 4 channels/iter.
// ------------------------------------------------------------------
__global__ __launch_bounds__(256) void k3_carafe(
    const float* __restrict__ X,   // (2,256,64,64)
    const float* __restrict__ Z,   // (2,100,64,64)
    float* __restrict__ Out)       // (2,256,128,128)
{
    __shared__ float tile[4][12 * 12];

    const int t   = threadIdx.x;
    const int pix = t >> 2;             // 0..63
    const int sub = t & 3;              // 0..3
    const int pi  = pix >> 3, pj = pix & 7;

    const int blk  = blockIdx.x;        // 0..127
    const int bimg = blk >> 6;
    const int tb   = blk & 63;
    const int th0  = (tb >> 3) * 8;
    const int tw0  = (tb & 7) * 8;

    const int hlo = th0 + pi, wlo = tw0 + pj;
    const int ys = sub >> 1, xs = sub & 1;
    const int y = 2 * hlo + ys, x = 2 * wlo + xs;

    // PixelShuffle gather + softmax over the 25 kernel logits (registers)
    const float* Zb = Z + bimg * (100 * 4096) + hlo * 64 + wlo;
    float wk[25];
    float mx = -1e30f;
    #pragma unroll
    for (int k = 0; k < 25; ++k) {
        wk[k] = Zb[(k * 4 + sub) * 4096];
        mx = fmaxf(mx, wk[k]);
    }
    float sum = 0.f;
    #pragma unroll
    for (int k = 0; k < 25; ++k) { wk[k] = __expf(wk[k] - mx); sum += wk[k]; }
    const float inv = 1.f / sum;
    #pragma unroll
    for (int k = 0; k < 25; ++k) wk[k] *= inv;

    const float* Xb = X + bimg * (256 * 4096);
    float* Ob = Out + bimg * (256 * 16384) + y * 128 + x;

    for (int c0 = 0; c0 < 256; c0 += 4) {
        __syncthreads();
        for (int idx = t; idx < 4 * 144; idx += 256) {
            const int ch = idx / 144;
            const int rem = idx - ch * 144;
            const int r = rem / 12;
            const int cc = rem - r * 12;
            const int hh = th0 - 2 + r, ww = tw0 - 2 + cc;
            float v = 0.f;
            if ((unsigned)hh < 64u && (unsigned)ww < 64u)
                v = Xb[(c0 + ch) * 4096 + hh * 64 + ww];
            tile[ch][r * 12 + cc] = v;
        }
        __syncthreads();
        #pragma unroll
        for (int ch = 0; ch < 4; ++ch) {
            float acc = 0.f;
            #pragma unroll
            for (int i = 0; i < 5; ++i)
                #pragma unroll
                for (int j = 0; j < 5; ++j)
                    acc += wk[i * 5 + j] * tile[ch][(pi + i) * 12 + (pj + j)];
            Ob[(c0 + ch) * 16384] = acc;
        }
    }
}

extern "C" void kernel_launch(void* const* d_in, const int* in_sizes, int n_in,
                              void* d_out, int out_size, void* d_ws, size_t ws_size,
                              hipStream_t stream) {
    const float* X   = (const float*)d_in[0];
    const float* Wc  = (const float*)d_in[1];
    const float* bc  = (const float*)d_in[2];
    const float* g1  = (const float*)d_in[3];
    const float* be1 = (const float*)d_in[4];
    const float* mu1 = (const float*)d_in[5];
    const float* va1 = (const float*)d_in[6];
    const float* We  = (const float*)d_in[7];
    const float* bev = (const float*)d_in[8];
    const float* g2  = (const float*)d_in[9];
    const float* be2 = (const float*)d_in[10];
    const float* mu2 = (const float*)d_in[11];
    const float* va2 = (const float*)d_in[12];

    float* Out = (float*)d_out;
    float* Y1  = (float*)d_ws;              // 8192*64 floats = 2.0 MB
    float* Z   = Y1 + 8192 * 64;            // 8192*100 floats = 3.1 MB

    k1_comp<<<256, 256, 0, stream>>>(X, Wc, bc, g1, be1, mu1, va1, Y1);
    k2_enc <<<448, 256, 0, stream>>>(Y1, We, bev, g2, be2, mu2, va2, Z);
    k3_carafe<<<128, 256, 0, stream>>>(X, Z, Out);
}